// YOLO_LOSS_68083821576347
// MI455X (gfx1250) — compile-verified
//
#include <hip/hip_runtime.h>
#include <hip/hip_bf16.h>
#include <stdint.h>

typedef __attribute__((ext_vector_type(2))) float v2f;
typedef __attribute__((ext_vector_type(8))) float v8f;

#define IMGSZ   416.0f
#define NCLS    5
#define EPSV    0.01f
#define L_NOOBJ 100.0f
#define L_COORD 100.0f
#define UNROLL  8

__device__ __forceinline__ float softplus_f(float x) {
    // stable log(1 + e^x)
    return fmaxf(x, 0.0f) + log1pf(__expf(-fabsf(x)));
}
__device__ __forceinline__ float sigmoid_f(float x) {
    return 1.0f / (1.0f + __expf(-x));
}

// ---------------------------------------------------------------------------
// Kernel 1: full = sum softplus(obj_map + EPS) over one scale.
// Streams ONLY channels (a*10+4) of the (B,30,N,N) tensor (1/10 of the bytes)
// via the CDNA5 async global->LDS path: 8 non-temporal async loads in flight
// per wave per iteration, one s_wait_asynccnt, then conflict-free LDS
// readback. Wave reduction uses a real v_wmma_f32_16x16x4_f32
// (A = partials 16x4 tile, B = ones).
// ---------------------------------------------------------------------------
__global__ void __launch_bounds__(256)
noobj_full_kernel(const float* __restrict__ y, int NN, int total, float* __restrict__ acc) {
    __shared__ float smem[UNROLL * 256];
    const int tid = threadIdx.x;
    const unsigned long long base = (unsigned long long)y;
    // Low 32 bits of a generic LDS pointer == LDS byte offset (ISA aperture rule).
    const uint32_t lds0 = (uint32_t)(uintptr_t)(&smem[tid]);

    float partial = 0.0f;
    const int stride = gridDim.x * blockDim.x;
    for (int e0 = blockIdx.x * blockDim.x + tid; e0 < total; e0 += stride * UNROLL) {
        // Issue up to 8 async DMA loads (ASYNCcnt), each EXEC-masked.
#pragma unroll
        for (int u = 0; u < UNROLL; ++u) {
            int e = e0 + u * stride;
            if (e < total) {
                // e -> (batch b, anchor a, cell off); obj channel = a*10 + 4
                int b   = e / (3 * NN);
                int r   = e - b * 3 * NN;
                int a   = r / NN;
                int off = r - a * NN;
                uint32_t byteoff = (uint32_t)(((b * 30 + a * 10 + 4) * NN + off) * 4);
                uint32_t ldsoff  = lds0 + (uint32_t)(u * 256 * 4);
                asm volatile("global_load_async_to_lds_b32 %0, %1, %2 th:TH_LOAD_NT"
                             :
                             : "v"(ldsoff), "v"(byteoff), "s"(base)
                             : "memory");
            }
        }
        // Drain this wave's async queue, then consume (own LDS slots only).
        asm volatile("s_wait_asynccnt 0" ::: "memory");
#pragma unroll
        for (int u = 0; u < UNROLL; ++u) {
            int e = e0 + u * stride;
            if (e < total) {
                float x = smem[u * 256 + tid];
                partial += softplus_f(x + EPSV);
            }
        }
    }

    // Wave reduction via WMMA: A[M][K] 16x4 holds the 32 lane partials
    // (lane L -> M=L%16, K = 2*(L/16); K odd slots zero), B = ones(4x16).
    // D[m][n] = partial[m] + partial[m+16]; lane n sums its 8 D VGPRs,
    // one shfl_xor(16) completes the 32-lane sum.
    v2f av = {partial, 0.0f};
    v2f bv = {1.0f, 1.0f};
    v8f cv = {0.0f, 0.0f, 0.0f, 0.0f, 0.0f, 0.0f, 0.0f, 0.0f};
    cv = __builtin_amdgcn_wmma_f32_16x16x4_f32(
        /*neg_a=*/false, av, /*neg_b=*/false, bv,
        /*c_mod=*/(short)0, cv, /*reuse_a=*/false, /*reuse_b=*/false);
    float s = cv[0] + cv[1] + cv[2] + cv[3] + cv[4] + cv[5] + cv[6] + cv[7];
    s += __shfl_xor(s, 16, 32);
    if ((tid & 31) == 0) atomicAdd(acc, s);
}

// ---------------------------------------------------------------------------
// Kernel 2: per-batch gathered-cell terms (IOU match + coord/obj/class/corr).
// One thread per batch; ~90 scattered loads each — tiny.
// ws layout: [0]=full [1]=corr [2]=lc*L_COORD [3]=lo [4]=lcl
// ---------------------------------------------------------------------------
__global__ void cell_terms_kernel(const float* __restrict__ y1,
                                  const float* __restrict__ y2,
                                  const float* __restrict__ y3,
                                  const float* __restrict__ t,
                                  float* __restrict__ ws, int B) {
    int b = blockIdx.x * blockDim.x + threadIdx.x;
    if (b >= B) return;

    const float anc[9][2] = {{30,61},{62,45},{59,119},{10,13},{16,30},
                             {33,23},{116,90},{156,198},{373,326}};

    float tx  = t[b*5+0], ty = t[b*5+1];
    float tws = t[b*5+2], ths = t[b*5+3];
    int   cls = (int)t[b*5+4];

    float tw = floorf(tws * IMGSZ), th = floorf(ths * IMGSZ);
    float iou[9];
    float best = -1e30f;
    for (int a = 0; a < 9; ++a) {
        float pw = anc[a][0], ph = anc[a][1];
        float A  = fmaxf(pw, tw) * fmaxf(ph, th);
        float I  = fminf(pw, tw) * fminf(ph, th);
        float TP = A - pw * ph - tw * th + 2.0f * I;
        iou[a]   = TP / (TP + A);
        best     = fmaxf(best, iou[a]);
    }
    float m[3];
    for (int a = 0; a < 3; ++a) m[a] = (iou[a] == best) ? 1.0f : 0.0f;

    const float* ys[3] = {y1, y2, y3};
    const int    Ns[3] = {13, 26, 52};

    float lc = 0.f, lo = 0.f, lcl = 0.f, corr = 0.f;
    for (int s = 0; s < 3; ++s) {
        int N = Ns[s], NN = N * N;
        int Cx = (int)floorf(tx * (float)N);
        int Cy = (int)floorf(ty * (float)N);
        float fx = tx * (float)N - (float)Cx;
        float fy = ty * (float)N - (float)Cy;
        int idx = Cy * N + Cx;
        const float* yb = ys[s] + (size_t)b * 30 * NN + idx;
        for (int a = 0; a < 3; ++a) {
            const float* ca = yb + a * 10 * NN;
            float c0 = ca[0], c1 = ca[NN], c2 = ca[2*NN], c3 = ca[3*NN], c4 = ca[4*NN];
            float sx = sigmoid_f(c0), sy = sigmoid_f(c1);
            float d0 = sx - fx, d1 = sy - fy, d2 = c2 - tws, d3 = c3 - ths;
            lc += m[a] * (d0*d0 + d1*d1 + d2*d2 + d3*d3);
            lo += m[a] * softplus_f(-c4);
            float bce = 0.f;
            for (int k = 0; k < NCLS; ++k) {
                float lg = ca[(5 + k) * NN];
                bce += softplus_f(lg) - lg * ((k == cls) ? 1.0f : 0.0f);
            }
            lcl  += m[a] * (bce * (1.0f / NCLS));
            corr += m[a] * logf(1.0f - sigmoid_f(c4) + EPSV);
        }
    }
    atomicAdd(&ws[1], corr);
    atomicAdd(&ws[2], L_COORD * lc);
    atomicAdd(&ws[3], lo);
    atomicAdd(&ws[4], lcl);
}

// ---------------------------------------------------------------------------
// Kernel 3: combine -> (loss, lc, lcl, ln, lo)
// ---------------------------------------------------------------------------
__global__ void finalize_kernel(const float* __restrict__ ws, float* __restrict__ out) {
    if (threadIdx.x == 0 && blockIdx.x == 0) {
        float full = ws[0], corr = ws[1];
        float lc = ws[2], lo = ws[3], lcl = ws[4];
        float ln = L_NOOBJ * (full + corr);
        out[0] = lc + lcl + ln + lo;
        out[1] = lc;
        out[2] = lcl;
        out[3] = ln;
        out[4] = lo;
    }
}

extern "C" void kernel_launch(void* const* d_in, const int* in_sizes, int n_in,
                              void* d_out, int out_size, void* d_ws, size_t ws_size,
                              hipStream_t stream) {
    const float* y1 = (const float*)d_in[0];   // (B,30,13,13)
    const float* y2 = (const float*)d_in[1];   // (B,30,26,26)
    const float* y3 = (const float*)d_in[2];   // (B,30,52,52)
    const float* t  = (const float*)d_in[3];   // (B,5)
    float* ws  = (float*)d_ws;
    float* out = (float*)d_out;

    const int B = in_sizes[3] / 5;             // 256

    hipMemsetAsync(d_ws, 0, 8 * sizeof(float), stream);

    const float* ys[3] = {y1, y2, y3};
    const int    Ns[3] = {13, 26, 52};
    for (int s = 0; s < 3; ++s) {
        int NN    = Ns[s] * Ns[s];
        int total = B * 3 * NN;                // only the 3 obj channels per batch
        int blocks = (total + 255) / 256;
        if (blocks > 1024) blocks = 1024;
        noobj_full_kernel<<<blocks, 256, 0, stream>>>(ys[s], NN, total, &ws[0]);
    }
    cell_terms_kernel<<<(B + 63) / 64, 64, 0, stream>>>(y1, y2, y3, t, ws, B);
    finalize_kernel<<<1, 1, 0, stream>>>(ws, out);
}